// SAGE_71399536328824
// MI455X (gfx1250) — compile-verified
//
#include <hip/hip_runtime.h>

// GraphSAGE, 10 layers, N=500K nodes, E=16M edges, widths 10->5->(5x8)->1.
// MI455X plan: all hot data fits in 192MB L2 (features <=20MB, edge list 128MB),
// so per-layer cost = edge gather/scatter served by L2. We:
//  1) transform-then-scatter (mean is linear): per-edge atomic floats 10->5
//     on layer 1 and 5->1 on layer 10,
//  2) hardware fp32 atomics (unsafeAtomicAdd -> global_atomic_add_f32) into
//     L2-resident agg,
//  3) degree reciprocals computed once, reused by all 10 layers,
//  4) node GEMMs via V_WMMA_F32_16X16X4_F32 (exact fp32), one wave per
//     16-node tile, agg-zeroing fused in. Fragment loads are branch-free
//     (clamped index + cndmask); stores use a wave-uniform full-tile fast
//     path (immediate-offset store burst, single exec-mask setup).

typedef __attribute__((ext_vector_type(2))) float v2f;
typedef __attribute__((ext_vector_type(8))) float v8f;

__global__ void zero_f32(float* __restrict__ p, long n) {
  long i = (long)blockIdx.x * blockDim.x + threadIdx.x;
  if (i < n) p[i] = 0.0f;
}

__global__ void count_edges(const int* __restrict__ dst, float* __restrict__ cnt,
                            long n_edges) {
  long i = (long)blockIdx.x * blockDim.x + threadIdx.x;
  if (i < n_edges) unsafeAtomicAdd(&cnt[dst[i]], 1.0f);
}

__global__ void reciprocal_clamped(float* __restrict__ p, long n) {
  long i = (long)blockIdx.x * blockDim.x + threadIdx.x;
  if (i < n) p[i] = 1.0f / fmaxf(p[i], 1.0f);
}

// One wave per 16-node tile. Via V_WMMA_F32_16X16X4_F32 computes
//   hl[node][:] = h[node][:] @ Wl.T         (scattered over edges next)
//   hs[node][:] = h[node][:] @ Wr.T + bias  (self term)
// and zeroes agg[node][:].
// A layout (32-bit 16x4): lanes 0-15 -> M=lane, K=k0,k0+1; lanes 16-31 ->
// same M, K=k0+2,k0+3. B (4x16) symmetric with column N = lane&15.
// D: vgpr v holds Y[v + 8*(lane>=16)][lane&15].
template <int F_IN, int F_OUT>
__global__ void node_transform(const float* __restrict__ h,
                               const float* __restrict__ Wl,
                               const float* __restrict__ Wr,
                               const float* __restrict__ bias,
                               float* __restrict__ hl,
                               float* __restrict__ hs,
                               float* __restrict__ agg,
                               int n_nodes) {
  const int lane = threadIdx.x & 31;
  const int wave = threadIdx.x >> 5;
  const int tile = blockIdx.x * (blockDim.x >> 5) + wave;
  const int tiles = (n_nodes + 15) >> 4;
  if (tile >= tiles) return;  // wave-uniform; EXEC stays all-ones for WMMA

  const int m    = lane & 15;   // A row / B,D column
  const int hi   = lane >> 4;   // K-half of the 4-slice
  const int row0 = tile << 4;
  const int arow = min(row0 + m, n_nodes - 1);
  const bool cok = (m < F_OUT);

  // Branch-free loads: clamped index + select (cndmask), never a branch.
  auto ldA = [&](int k) -> float {
    const bool ok = (k < F_IN);
    const float v = h[(long)arow * F_IN + (ok ? k : 0)];
    return ok ? v : 0.0f;
  };
  auto ldW = [&](const float* __restrict__ W, int k) -> float {
    const bool ok = cok && (k < F_IN);
    const float v = W[ok ? (m * F_IN + k) : 0];
    return ok ? v : 0.0f;
  };

  v8f dl = {};
  v8f dr = {};
#pragma unroll
  for (int kc = 0; kc < F_IN; kc += 4) {
    const int k0 = kc + hi * 2;
    v2f a, bl, br;
    a.x  = ldA(k0);
    a.y  = ldA(k0 + 1);
    bl.x = ldW(Wl, k0);
    bl.y = ldW(Wl, k0 + 1);
    br.x = ldW(Wr, k0);
    br.y = ldW(Wr, k0 + 1);
    // 8 args: (neg_a, A, neg_b, B, c_mod, C, reuse_a, reuse_b)
    dl = __builtin_amdgcn_wmma_f32_16x16x4_f32(false, a, false, bl, (short)0,
                                               dl, false, false);
    dr = __builtin_amdgcn_wmma_f32_16x16x4_f32(false, a, false, br, (short)0,
                                               dr, false, false);
  }

  // Epilogue: single exec-mask setup (cok); wave-uniform full-tile fast path
  // lets all 8 node-stores use one base address + immediate offsets.
  if (cok) {
    const float bv   = bias[m];
    const long  base = (long)(row0 + hi * 8) * F_OUT + m;
    if (row0 + 16 <= n_nodes) {  // full tile (always true when N % 16 == 0)
#pragma unroll
      for (int v = 0; v < 8; ++v) {
        hl[base + (long)v * F_OUT]  = dl[v];
        hs[base + (long)v * F_OUT]  = dr[v] + bv;
        agg[base + (long)v * F_OUT] = 0.0f;
      }
    } else {  // ragged last tile
#pragma unroll
      for (int v = 0; v < 8; ++v) {
        if (row0 + hi * 8 + v < n_nodes) {
          hl[base + (long)v * F_OUT]  = dl[v];
          hs[base + (long)v * F_OUT]  = dr[v] + bv;
          agg[base + (long)v * F_OUT] = 0.0f;
        }
      }
    }
  }
}

// Edge-parallel scatter of transformed features; HW fp32 atomics, L2-resident.
template <int F>
__global__ void edge_scatter(const int* __restrict__ src,
                             const int* __restrict__ dst,
                             const float* __restrict__ hl,
                             float* __restrict__ agg, long n_edges) {
  long i = (long)blockIdx.x * blockDim.x + threadIdx.x;
  if (i >= n_edges) return;
  const int s = src[i];
  const int d = dst[i];
  float v[F];
#pragma unroll
  for (int j = 0; j < F; ++j) v[j] = hl[(long)s * F + j];
#pragma unroll
  for (int j = 0; j < F; ++j) unsafeAtomicAdd(&agg[(long)d * F + j], v[j]);
}

// out[node][:] = [relu](agg * inv_degree + self); one thread per node.
template <int F, bool RELU>
__global__ void combine(const float* __restrict__ agg,
                        const float* __restrict__ hs,
                        const float* __restrict__ inv,
                        float* __restrict__ out, int n_nodes) {
  const int node = blockIdx.x * blockDim.x + threadIdx.x;
  if (node >= n_nodes) return;
  const float iv = inv[node];
#pragma unroll
  for (int j = 0; j < F; ++j) {
    const long o = (long)node * F + j;
    const float v = agg[o] * iv + hs[o];
    out[o] = RELU ? fmaxf(v, 0.0f) : v;
  }
}

extern "C" void kernel_launch(void* const* d_in, const int* in_sizes, int n_in,
                              void* d_out, int out_size, void* d_ws,
                              size_t ws_size, hipStream_t stream) {
  const float* x    = (const float*)d_in[0];   // [N,10]
  const int*   ei   = (const int*)  d_in[1];   // [2,E]
  const float* Wl1  = (const float*)d_in[2];   // [5,10]
  const float* Wr1  = (const float*)d_in[3];   // [5,10]
  const float* b1   = (const float*)d_in[4];   // [5]
  const float* Wlm  = (const float*)d_in[5];   // [8,5,5]
  const float* Wrm  = (const float*)d_in[6];   // [8,5,5]
  const float* bm   = (const float*)d_in[7];   // [8,5]
  const float* Wl10 = (const float*)d_in[8];   // [1,5]
  const float* Wr10 = (const float*)d_in[9];   // [1,5]
  const float* b10  = (const float*)d_in[10];  // [1]

  const int  N = in_sizes[0] / 10;
  const long E = (long)in_sizes[1] / 2;
  const int* src = ei;
  const int* dst = ei + E;

  // Workspace (floats): inv[N] | hl[N*5] | hs[N*5] | agg[N*5] | hA[N*5] | hB[N*5]
  float* ws  = (float*)d_ws;
  float* inv = ws;
  float* hl  = inv + (size_t)N;
  float* hs  = hl + (size_t)N * 5;
  float* agg = hs + (size_t)N * 5;
  float* hA  = agg + (size_t)N * 5;
  float* hB  = hA + (size_t)N * 5;

  const int B = 256;
  auto cdiv = [](long a, long b) { return (int)((a + b - 1) / b); };

  // Degree reciprocals (shared by all layers).
  zero_f32<<<cdiv(N, B), B, 0, stream>>>(inv, N);
  count_edges<<<cdiv(E, B), B, 0, stream>>>(dst, inv, E);
  reciprocal_clamped<<<cdiv(N, B), B, 0, stream>>>(inv, N);

  const int tiles   = (N + 15) / 16;
  const int tblocks = (tiles + 7) / 8;  // 8 wave32 waves per 256-thread block

  // Layer 1: 10 -> 5, ReLU
  node_transform<10, 5><<<tblocks, 256, 0, stream>>>(x, Wl1, Wr1, b1, hl, hs, agg, N);
  edge_scatter<5><<<cdiv(E, B), B, 0, stream>>>(src, dst, hl, agg, E);
  combine<5, true><<<cdiv(N, B), B, 0, stream>>>(agg, hs, inv, hA, N);

  // Layers 2..9: 5 -> 5, ReLU (ping-pong hA/hB; ends in hA after 8 steps)
  for (int l = 0; l < 8; ++l) {
    const float* hin  = (l & 1) ? hB : hA;
    float*       hout = (l & 1) ? hA : hB;
    node_transform<5, 5><<<tblocks, 256, 0, stream>>>(
        hin, Wlm + l * 25, Wrm + l * 25, bm + l * 5, hl, hs, agg, N);
    edge_scatter<5><<<cdiv(E, B), B, 0, stream>>>(src, dst, hl, agg, E);
    combine<5, true><<<cdiv(N, B), B, 0, stream>>>(agg, hs, inv, hout, N);
  }

  // Layer 10: 5 -> 1, no ReLU; only ONE float scattered per edge.
  node_transform<5, 1><<<tblocks, 256, 0, stream>>>(hA, Wl10, Wr10, b10, hl, hs, agg, N);
  edge_scatter<1><<<cdiv(E, B), B, 0, stream>>>(src, dst, hl, agg, E);
  combine<1, false><<<cdiv(N, B), B, 0, stream>>>(agg, hs, inv, (float*)d_out, N);
}